// SSKernel_730144440570
// MI455X (gfx1250) — compile-verified
//
#include <hip/hip_runtime.h>
#include <hip/hip_bf16.h>

typedef __attribute__((ext_vector_type(2))) float v2f;
typedef __attribute__((ext_vector_type(8))) float v8f;

namespace {
constexpr int kH  = 1024;   // heads
constexpr int kN  = 32;     // half conjugate-pair modes
constexpr int kL  = 2048;   // sequence length
constexpr int kLf = kL / 2 + 1;   // 1025 rfft bins
constexpr int kKH = 1040;   // padded half-K (>= kLf, multiple of 16)
constexpr int kK  = 2 * kKH; // GEMM K = 2080 (multiple of 16)
}

// ---------------------------------------------------------------------------
// CDNA5 async staging helpers (GLOBAL_LOAD_ASYNC_TO_LDS_*, ASYNCcnt).
// GVS mode: mem addr = SGPR base + 32-bit per-lane offset; per-lane LDS
// destination byte address in a VGPR (low 32 bits of a generic LDS pointer).
// ---------------------------------------------------------------------------
__device__ __forceinline__ void async_ld_b128(uint32_t lds_addr, uint32_t voff,
                                              const void* sbase) {
  asm volatile("global_load_async_to_lds_b128 %0, %1, %2 offset:0"
               :: "v"(lds_addr), "v"(voff), "s"(sbase)
               : "memory");
}
__device__ __forceinline__ void wait_async0() {
  asm volatile("s_wait_asynccnt 0" ::: "memory");
}

// ---------------------------------------------------------------------------
// Kernel 1: Cauchy reduction + Woodbury + bilinear factor -> kf (complex),
// written as [Re | Im] rows of the GEMM A matrix (H x kK), zero padded.
// z_l = 2i*tan(pi*l/L); 2/(1+omega_l) = 1 + i*tan(pi*l/L). Nyquist handled
// via its analytic limit kf = dt * Re(sum_n B_n*C_n).
// ---------------------------------------------------------------------------
__global__ __launch_bounds__(256) void ssm_cauchy_kf(
    const float* __restrict__ log_dt, const float* __restrict__ w,
    const float* __restrict__ Bv, const float* __restrict__ Cv,
    const float* __restrict__ Pv, float* __restrict__ Abuf)
{
  __shared__ float swr[kN], swi[kN];
  __shared__ float s00r[kN], s00i[kN];   // B*C
  __shared__ float s01r[kN], s01i[kN];   // B*conj(P)
  __shared__ float s10r[kN], s10i[kN];   // P*C
  __shared__ float s11r[kN];             // P*conj(P) = |P|^2 (real)
  __shared__ float s_nyq;

  const int h = blockIdx.x;
  const int tid = threadIdx.x;
  const float dt = expf(log_dt[h]);

  if (tid < kN) {
    const int n = tid;
    const int base = (h * kN + n) * 2;
    const float wr = w[base] * dt, wi = w[base + 1] * dt;   // w * dt
    swr[n] = wr; swi[n] = wi;
    const float Br = Bv[base], Bi = Bv[base + 1];
    const float Cr = Cv[base], Ci = Cv[base + 1];           // CH = 1
    const float Pr = Pv[base], Pi = Pv[base + 1];           // R  = 1
    s00r[n] = Br * Cr - Bi * Ci;  s00i[n] = Br * Ci + Bi * Cr;
    s01r[n] = Br * Pr + Bi * Pi;  s01i[n] = Bi * Pr - Br * Pi;
    s10r[n] = Pr * Cr - Pi * Ci;  s10i[n] = Pr * Ci + Pi * Cr;
    s11r[n] = Pr * Pr + Pi * Pi;
  }
  __syncthreads();
  if (tid == 0) {
    float s = 0.f;
    for (int n = 0; n < kN; ++n) s += s00r[n];
    s_nyq = s * dt;                       // z -> inf limit of kf
  }
  __syncthreads();

  for (int l = tid; l < kKH; l += 256) {
    float kr, ki;
    if (l > kL / 2) {                     // zero padding columns
      kr = 0.f; ki = 0.f;
    } else if (l == kL / 2) {             // Nyquist: analytic limit (real)
      kr = s_nyq; ki = 0.f;
    } else {
      const float x  = (float)l * (1.0f / (float)kL);  // in [0, 0.5)
      const float t  = sinpif(x) / cospif(x);          // tan(pi*l/L)
      const float t2 = 2.f * t;                        // Im(z)
      float r00r = 0.f, r00i = 0.f, r01r = 0.f, r01i = 0.f;
      float r10r = 0.f, r10i = 0.f, r11r = 0.f, r11i = 0.f;
      #pragma unroll 4
      for (int n = 0; n < kN; ++n) {
        const float wr = swr[n], wi = swi[n];
        const float e1 = t2 - wi, e2 = t2 + wi;
        const float inv1 = 1.0f / (wr * wr + e1 * e1);
        const float inv2 = 1.0f / (wr * wr + e2 * e2);
        const float r1r = -wr * inv1, r1i = -e1 * inv1;  // 1/(z - w)
        const float r2r = -wr * inv2, r2i = -e2 * inv2;  // 1/(z - conj(w))
        const float sr = r1r + r2r, dr = r1r - r2r;
        const float si = r1i + r2i, di = r1i - r2i;
        float vr, vi;
        vr = s00r[n]; vi = s00i[n];
        r00r += vr * sr - vi * di;  r00i += vr * si + vi * dr;
        vr = s01r[n]; vi = s01i[n];
        r01r += vr * sr - vi * di;  r01i += vr * si + vi * dr;
        vr = s10r[n]; vi = s10i[n];
        r10r += vr * sr - vi * di;  r10i += vr * si + vi * dr;
        vr = s11r[n];
        r11r += vr * sr;            r11i += vr * si;
      }
      r00r *= dt; r00i *= dt; r01r *= dt; r01i *= dt;
      r10r *= dt; r10i *= dt; r11r *= dt; r11i *= dt;
      // Woodbury rank-1 correction: kf = r00 - r01*r10/(1+r11)
      const float ddr = 1.f + r11r, ddi = r11i;
      const float dinv = 1.0f / (ddr * ddr + ddi * ddi);
      const float numr = r01r * r10r - r01i * r10i;
      const float numi = r01r * r10i + r01i * r10r;
      const float cr = (numr * ddr + numi * ddi) * dinv;
      const float ci = (numi * ddr - numr * ddi) * dinv;
      const float fr = r00r - cr, fi = r00i - ci;
      kr = fr - fi * t;                   // * (1 + i*t) == * 2/(1+omega)
      ki = fi + fr * t;
    }
    Abuf[(size_t)h * kK + l]       = kr;
    Abuf[(size_t)h * kK + kKH + l] = ki;
  }
}

// ---------------------------------------------------------------------------
// Kernel 2: inverse-rDFT weight matrix (kK x kL).
// Rows [0,kLf):        scale_l *  cos(2*pi*l*t/L)
// Rows [kKH,kKH+kLf): -scale_l *  sin(2*pi*l*t/L)     (others zero)
// scale = 1/L for l in {0, L/2}, else 2/L. Phase reduced exactly mod L.
// ---------------------------------------------------------------------------
__global__ __launch_bounds__(256) void ssm_idft_w(float* __restrict__ Wbuf)
{
  const int t = blockIdx.x * 256 + threadIdx.x;   // 0..kL-1
  const int r = blockIdx.y;                       // 0..kK-1
  const int l = (r < kKH) ? r : (r - kKH);
  float val = 0.f;
  if (l <= kL / 2) {
    const int   m  = (l * t) & (kL - 1);          // exact phase reduction
    const float ph = (float)m * (2.0f / (float)kL);
    const float scale = (l == 0 || l == kL / 2) ? (1.0f / (float)kL)
                                                : (2.0f / (float)kL);
    val = (r < kKH) ? scale * cospif(ph) : -scale * sinpif(ph);
  }
  Wbuf[(size_t)r * kL + t] = val;
}

// ---------------------------------------------------------------------------
// Kernel 3: fp32 WMMA GEMM  out(1024x2048) = A(1024x2080) * W(2080x2048)
// 256 threads = 8 waves; block tile 64x128; wave tile 32x32 (2x2 WMMA tiles,
// every fragment reused twice); BK=16, double-buffered LDS filled by
// GLOBAL_LOAD_ASYNC_TO_LDS_B128 (ASYNCcnt) so the next K-chunk streams in
// while WMMAs consume the current one. LDS strides 20/136 keep the two
// half-wave lane groups on disjoint subsets of the 64 banks.
// ---------------------------------------------------------------------------
__global__ __launch_bounds__(256) void ssm_idft_gemm(
    const float* __restrict__ A, const float* __restrict__ W,
    float* __restrict__ out)
{
  __shared__ float As[2][64][20];      // 10.0 KB
  __shared__ float Bs[2][16][136];     // 17.0 KB

  const int tid  = threadIdx.x;
  const int lane = tid & 31;
  const int wave = tid >> 5;
  const int wm   = wave & 1;             // wave row (32 rows each)
  const int wn   = wave >> 1;            // wave col (32 cols each)
  const int m16  = lane & 15;
  const int khi  = (lane >> 4) << 1;     // 0 or 2: K-subgroup per ISA layout
  const int row0 = blockIdx.y * 64;
  const int col0 = blockIdx.x * 128;

  // async staging geometry: one b128 per lane for A, two for B, per chunk
  const int arow = tid >> 2, acol = (tid & 3) << 2;    // A chunk 64x16
  const int brow = tid >> 5, bcol = (tid & 31) << 2;   // B chunk 16x128
  const uint32_t ldsA[2]  = { (uint32_t)(uintptr_t)&As[0][arow][acol],
                              (uint32_t)(uintptr_t)&As[1][arow][acol] };
  const uint32_t ldsBa[2] = { (uint32_t)(uintptr_t)&Bs[0][brow][bcol],
                              (uint32_t)(uintptr_t)&Bs[1][brow][bcol] };
  const uint32_t ldsBb[2] = { (uint32_t)(uintptr_t)&Bs[0][brow + 8][bcol],
                              (uint32_t)(uintptr_t)&Bs[1][brow + 8][bcol] };
  const uint32_t aoff0 = (uint32_t)(((row0 + arow) * kK + acol) * 4);
  const uint32_t boff0 = (uint32_t)((brow * kL + col0 + bcol) * 4);
  const uint32_t boff1 = boff0 + 8u * (uint32_t)kL * 4u;

  // preload chunk 0 into buffer 0
  async_ld_b128(ldsA[0],  aoff0, A);
  async_ld_b128(ldsBa[0], boff0, W);
  async_ld_b128(ldsBb[0], boff1, W);

  v8f acc00 = {}, acc01 = {}, acc10 = {}, acc11 = {};

  int cur = 0;
  for (int kb = 0; kb < kK; kb += 16) {
    wait_async0();          // my chunk-cur transfers have landed in LDS
    __syncthreads();        // all waves loaded cur & finished reading cur^1
    const int nkb = kb + 16;
    if (nkb < kK) {         // stream next chunk into the other buffer
      const int nxt = cur ^ 1;
      const uint32_t ka = aoff0 + (uint32_t)(nkb * 4);
      const uint32_t kb4 = (uint32_t)(nkb) * (uint32_t)kL * 4u;
      async_ld_b128(ldsA[nxt],  ka,          A);
      async_ld_b128(ldsBa[nxt], boff0 + kb4, W);
      async_ld_b128(ldsBb[nxt], boff1 + kb4, W);
    }
    #pragma unroll
    for (int kk = 0; kk < 16; kk += 4) {
      // A fragments 16x4: lane holds (m = m16, K = khi, khi+1)
      v2f aLo, aHi;
      aLo.x = As[cur][wm * 32 + m16][kk + khi];
      aLo.y = As[cur][wm * 32 + m16][kk + khi + 1];
      aHi.x = As[cur][wm * 32 + 16 + m16][kk + khi];
      aHi.y = As[cur][wm * 32 + 16 + m16][kk + khi + 1];
      // B fragments 4x16: lane holds (n = m16, K = khi, khi+1)
      const int cb = wn * 32 + m16;
      v2f b0, b1;
      b0.x = Bs[cur][kk + khi][cb];        b0.y = Bs[cur][kk + khi + 1][cb];
      b1.x = Bs[cur][kk + khi][cb + 16];   b1.y = Bs[cur][kk + khi + 1][cb + 16];
      acc00 = __builtin_amdgcn_wmma_f32_16x16x4_f32(false, aLo, false, b0,
                                                    (short)0, acc00, false, false);
      acc01 = __builtin_amdgcn_wmma_f32_16x16x4_f32(false, aLo, false, b1,
                                                    (short)0, acc01, false, false);
      acc10 = __builtin_amdgcn_wmma_f32_16x16x4_f32(false, aHi, false, b0,
                                                    (short)0, acc10, false, false);
      acc11 = __builtin_amdgcn_wmma_f32_16x16x4_f32(false, aHi, false, b1,
                                                    (short)0, acc11, false, false);
    }
    cur ^= 1;
  }

  // D layout: VGPR r holds M = r + 8*(lane>=16), N = lane&15
  const int orow = row0 + wm * 32 + ((lane >> 4) << 3);
  const int ocol = col0 + wn * 32 + m16;
  #pragma unroll
  for (int r = 0; r < 8; ++r) {
    out[(size_t)(orow + r) * kL + ocol]           = acc00[r];
    out[(size_t)(orow + r) * kL + ocol + 16]      = acc01[r];
    out[(size_t)(orow + 16 + r) * kL + ocol]      = acc10[r];
    out[(size_t)(orow + 16 + r) * kL + ocol + 16] = acc11[r];
  }
}

// ---------------------------------------------------------------------------
extern "C" void kernel_launch(void* const* d_in, const int* in_sizes, int n_in,
                              void* d_out, int out_size, void* d_ws, size_t ws_size,
                              hipStream_t stream) {
  (void)in_sizes; (void)n_in; (void)out_size; (void)ws_size;
  const float* log_dt = (const float*)d_in[0];
  const float* w      = (const float*)d_in[1];
  const float* B      = (const float*)d_in[2];
  const float* C      = (const float*)d_in[3];
  const float* P      = (const float*)d_in[4];
  // d_in[5] is L (=2048), baked into the kernels as a constant.
  float* out  = (float*)d_out;                        // (CH=1, H, L) fp32
  float* Abuf = (float*)d_ws;                         // H x kK   (8.5 MB)
  float* Wbuf = Abuf + (size_t)kH * kK;               // kK x kL (17.0 MB)

  ssm_cauchy_kf<<<kH, 256, 0, stream>>>(log_dt, w, B, C, P, Abuf);
  ssm_idft_w<<<dim3(kL / 256, kK), 256, 0, stream>>>(Wbuf);
  ssm_idft_gemm<<<dim3(kL / 128, kH / 64), 256, 0, stream>>>(Abuf, Wbuf, out);
}